// ELPH_24309514895388
// MI455X (gfx1250) — compile-verified
//
#include <hip/hip_runtime.h>
#include <math.h>

#define N_NODES 50000
#define N_EDGES 800000
#define F_INDIM 128
#define HDIM 64
#define NLAYERS 2
#define NUM_PERM 128
#define HLL_M 64

typedef float v2f __attribute__((ext_vector_type(2)));
typedef float v8f __attribute__((ext_vector_type(8)));

// ---------------------------------------------------------------------------
// WMMA f32 GEMM: C[m,n] = sum_k A[m,k] * W[k,n] (+ bias[n])
// A: [M x K] row-major, W: [K x HDIM] row-major, C: [M x HDIM].
// grid.x = M/16, blockDim = 128 (4 waves); wave w owns n-tile w (HDIM/16 = 4).
// Uses V_WMMA_F32_16X16X4_F32 per the CDNA5 ISA layouts:
//   A 16x4 : lane l -> row m = l&15, vgpr v -> k = 2*(l>>4) + v
//   B 4x16 : lane l -> col n = l&15, vgpr v -> k = 2*(l>>4) + v
//   C 16x16: lane l -> col n = l&15, vgpr v -> row m = v + 8*(l>>4)
// ---------------------------------------------------------------------------
__global__ void gemm_f32_wmma(const float* __restrict__ A,
                              const float* __restrict__ W,
                              const float* __restrict__ bias,
                              float* __restrict__ C,
                              int K) {
  const int lane = threadIdx.x & 31;
  const int wave = threadIdx.x >> 5;          // n-tile index 0..3
  const int hi   = lane >> 4;                 // lane group 0/1
  const int m0   = blockIdx.x * 16;
  const int n    = (wave << 4) + (lane & 15);
  const int mA   = m0 + (lane & 15);

  v8f acc = {};
  for (int k0 = 0; k0 < K; k0 += 4) {
    const int ka = k0 + (hi << 1);
    v2f a, b;
    a.x = A[(size_t)mA * K + ka];
    a.y = A[(size_t)mA * K + ka + 1];
    b.x = W[(size_t)ka * HDIM + n];
    b.y = W[(size_t)(ka + 1) * HDIM + n];
    acc = __builtin_amdgcn_wmma_f32_16x16x4_f32(
        /*neg_a=*/false, a, /*neg_b=*/false, b,
        /*c_mod=*/(short)0, acc, /*reuse_a=*/false, /*reuse_b=*/false);
  }
  const float bn = bias ? bias[n] : 0.0f;
#pragma unroll
  for (int v = 0; v < 8; ++v) {
    const int m = m0 + v + (hi << 3);
    C[(size_t)m * HDIM + n] = acc[v] + bn;
  }
}

// ---------------------------------------------------------------------------
// Degree / norm helpers
// ---------------------------------------------------------------------------
__global__ void k_deg_init(int* __restrict__ deg) {
  int i = blockIdx.x * blockDim.x + threadIdx.x;
  if (i < N_NODES) deg[i] = 1;  // self loop
}

__global__ void k_deg_count(const int* __restrict__ dst, int* __restrict__ deg) {
  int e = blockIdx.x * blockDim.x + threadIdx.x;
  if (e < N_EDGES) atomicAdd(&deg[dst[e]], 1);
}

__global__ void k_dis(const int* __restrict__ deg, float* __restrict__ dis) {
  int i = blockIdx.x * blockDim.x + threadIdx.x;
  if (i < N_NODES) dis[i] = rsqrtf((float)deg[i]);
}

// acc[i,f] = h[i,f] + b[f] + t[i,f]*dis[i]^2   (residual + bias + self-loop msg)
__global__ void k_init_acc(const float* __restrict__ h, const float* __restrict__ t,
                           const float* __restrict__ dis, const float* __restrict__ b,
                           float* __restrict__ acc) {
  int i = blockIdx.x * blockDim.x + threadIdx.x;
  if (i < N_NODES * HDIM) {
    int node = i >> 6;
    int f = i & 63;
    float dn = dis[node];
    acc[i] = h[i] + b[f] + t[i] * dn * dn;
  }
}

// ---------------------------------------------------------------------------
// One wave per edge: GCN scatter-add (2 f32/lane), HLL scatter-max (2 i32/lane),
// minhash scatter-min (4 i32/lane). Atomics resolve in L2 (192MB, holds the
// whole working set).
// ---------------------------------------------------------------------------
__global__ void k_edge_prop(const int* __restrict__ src, const int* __restrict__ dst,
                            const float* __restrict__ dis, const float* __restrict__ t,
                            float* __restrict__ acc,
                            const int* __restrict__ hll_old, int* __restrict__ hll_new,
                            const int* __restrict__ mh_old, int* __restrict__ mh_new) {
  const int gid = blockIdx.x * blockDim.x + threadIdx.x;
  const int e = gid >> 5;
  const int lane = gid & 31;
  if (e >= N_EDGES) return;
  const int s = src[e];
  const int d = dst[e];
  const float nrm = dis[s] * dis[d];

  const size_t sh = (size_t)s * HDIM, dh = (size_t)d * HDIM;
  atomicAdd(&acc[dh + lane], t[sh + lane] * nrm);
  atomicAdd(&acc[dh + lane + 32], t[sh + lane + 32] * nrm);

  const size_t sr = (size_t)s * HLL_M, dr = (size_t)d * HLL_M;
  atomicMax(&hll_new[dr + lane], hll_old[sr + lane]);
  atomicMax(&hll_new[dr + lane + 32], hll_old[sr + lane + 32]);

  const size_t sm = (size_t)s * NUM_PERM, dm = (size_t)d * NUM_PERM;
#pragma unroll
  for (int j = 0; j < 4; ++j)
    atomicMin(&mh_new[dm + lane + 32 * j], mh_old[sm + lane + 32 * j]);
}

// cards[n, layer] = alpha * m^2 / sum_j 2^-reg[j]
__global__ void k_cards(const int* __restrict__ hll, float* __restrict__ cards_out,
                        int layer) {
  int n = blockIdx.x * blockDim.x + threadIdx.x;
  if (n >= N_NODES) return;
  const int* r = hll + (size_t)n * HLL_M;
  float s = 0.0f;
#pragma unroll 8
  for (int j = 0; j < HLL_M; ++j) s += exp2f(-(float)r[j]);
  const float alpha = 0.7213f / (1.0f + 1.079f / (float)HLL_M);
  cards_out[(size_t)n * NLAYERS + layer] = alpha * (float)(HLL_M * HLL_M) / s;
}

__global__ void k_copy_f32(const float* __restrict__ in, float* __restrict__ out, int n) {
  int i = blockIdx.x * blockDim.x + threadIdx.x;
  if (i < n) out[i] = in[i];
}

__global__ void k_i32_to_f32(const int* __restrict__ in, float* __restrict__ out, int n) {
  int i = blockIdx.x * blockDim.x + threadIdx.x;
  if (i < n) out[i] = (float)in[i];
}

// ---------------------------------------------------------------------------
extern "C" void kernel_launch(void* const* d_in, const int* in_sizes, int n_in,
                              void* d_out, int out_size, void* d_ws, size_t ws_size,
                              hipStream_t stream) {
  (void)in_sizes; (void)n_in; (void)out_size; (void)ws_size;

  const float* x       = (const float*)d_in[0];
  const int*   ei      = (const int*)d_in[1];
  const int*   src     = ei;              // edge_index[0]
  const int*   dst     = ei + N_EDGES;    // edge_index[1]
  const int*   mh0     = (const int*)d_in[2];
  const int*   hll0    = (const int*)d_in[3];
  const float* W_enc   = (const float*)d_in[4];
  const float* b_enc   = (const float*)d_in[5];
  const float* W_convs = (const float*)d_in[6];   // [L, H, H]
  const float* b_convs = (const float*)d_in[7];   // [L, H]

  // workspace carve-out
  char* ws = (char*)d_ws;
  size_t off = 0;
  auto carve = [&](size_t bytes) -> void* {
    void* p = ws + off;
    off += (bytes + 255) & ~(size_t)255;
    return p;
  };
  int*   deg  = (int*)carve((size_t)N_NODES * 4);
  float* dis  = (float*)carve((size_t)N_NODES * 4);
  float* h0   = (float*)carve((size_t)N_NODES * HDIM * 4);
  float* h1   = (float*)carve((size_t)N_NODES * HDIM * 4);
  float* tb   = (float*)carve((size_t)N_NODES * HDIM * 4);
  int*   hllA = (int*)carve((size_t)N_NODES * HLL_M * 4);
  int*   hllB = (int*)carve((size_t)N_NODES * HLL_M * 4);
  int*   mhA  = (int*)carve((size_t)N_NODES * NUM_PERM * 4);
  int*   mhB  = (int*)carve((size_t)N_NODES * NUM_PERM * 4);

  // output regions (concatenated: h, cards, mh, hll)
  float* out_h     = (float*)d_out;
  float* out_cards = out_h + (size_t)N_NODES * HDIM;
  float* out_mh    = out_cards + (size_t)N_NODES * NLAYERS;
  float* out_hll   = out_mh + (size_t)N_NODES * NUM_PERM;

  const int BLK = 256;
  const int nodeBlocks  = (N_NODES + BLK - 1) / BLK;
  const int edgeBlocks  = (N_EDGES + BLK - 1) / BLK;
  const int featBlocks  = (N_NODES * HDIM + BLK - 1) / BLK;
  const int waveEdgeBlk = (N_EDGES * 32 + BLK - 1) / BLK;  // wave per edge
  const int gemmGrid    = N_NODES / 16;                    // 3125, exact

  // 1) symmetric GCN norm: dis = rsqrt(indeg + 1)
  k_deg_init<<<nodeBlocks, BLK, 0, stream>>>(deg);
  k_deg_count<<<edgeBlocks, BLK, 0, stream>>>(dst, deg);
  k_dis<<<nodeBlocks, BLK, 0, stream>>>(deg, dis);

  // 2) encoder: h0 = x @ W_enc + b_enc   (WMMA f32, K=128)
  gemm_f32_wmma<<<gemmGrid, 128, 0, stream>>>(x, W_enc, b_enc, h0, F_INDIM);

  // 3) layer 0
  gemm_f32_wmma<<<gemmGrid, 128, 0, stream>>>(h0, W_convs, nullptr, tb, HDIM);
  k_init_acc<<<featBlocks, BLK, 0, stream>>>(h0, tb, dis, b_convs, h1);
  hipMemcpyAsync(hllA, hll0, (size_t)N_NODES * HLL_M * 4, hipMemcpyDeviceToDevice, stream);
  hipMemcpyAsync(mhA, mh0, (size_t)N_NODES * NUM_PERM * 4, hipMemcpyDeviceToDevice, stream);
  k_edge_prop<<<waveEdgeBlk, BLK, 0, stream>>>(src, dst, dis, tb, h1,
                                               hll0, hllA, mh0, mhA);
  k_cards<<<nodeBlocks, BLK, 0, stream>>>(hllA, out_cards, 0);

  // 4) layer 1
  gemm_f32_wmma<<<gemmGrid, 128, 0, stream>>>(h1, W_convs + HDIM * HDIM, nullptr, tb, HDIM);
  k_init_acc<<<featBlocks, BLK, 0, stream>>>(h1, tb, dis, b_convs + HDIM, h0);
  hipMemcpyAsync(hllB, hllA, (size_t)N_NODES * HLL_M * 4, hipMemcpyDeviceToDevice, stream);
  hipMemcpyAsync(mhB, mhA, (size_t)N_NODES * NUM_PERM * 4, hipMemcpyDeviceToDevice, stream);
  k_edge_prop<<<waveEdgeBlk, BLK, 0, stream>>>(src, dst, dis, tb, h0,
                                               hllA, hllB, mhA, mhB);
  k_cards<<<nodeBlocks, BLK, 0, stream>>>(hllB, out_cards, 1);

  // 5) finalize outputs
  k_copy_f32<<<featBlocks, BLK, 0, stream>>>(h0, out_h, N_NODES * HDIM);
  k_i32_to_f32<<<(N_NODES * NUM_PERM + BLK - 1) / BLK, BLK, 0, stream>>>(
      mhB, out_mh, N_NODES * NUM_PERM);
  k_i32_to_f32<<<(N_NODES * HLL_M + BLK - 1) / BLK, BLK, 0, stream>>>(
      hllB, out_hll, N_NODES * HLL_M);
}